// Encoder_88596585382407
// MI455X (gfx1250) — compile-verified
//
#include <hip/hip_runtime.h>

// Instant-NGP style multi-resolution hash-grid encoder for MI455X (gfx1250).
//
// Mapping: one thread per (point, level). 1M points x 16 levels = 16M threads,
// wave32, 8 waves / 256-thread block. Each thread:
//   - loads x[p] (12B, L1-cached, 16x reuse within a half-wave)
//   - computes 8 corner hashes in uint32 (low 19 bits == reference int64 math)
//   - issues all 8 float2 gathers up front (L2-resident: 64MB table < 192MB L2)
//   - trilinear blend, one coalesced float2 store (wave stores 256B contiguous)

namespace {
constexpr int      kLevels   = 16;
constexpr int      kFeats    = 2;
constexpr unsigned kHashSize = 1u << 19;
constexpr unsigned kHashMask = kHashSize - 1u;
constexpr unsigned kP0       = 1958374283u;
constexpr unsigned kP1       = 2654435761u;
constexpr unsigned kP2       = 805459861u;
}  // namespace

__global__ __launch_bounds__(256) void hashgrid_encode_kernel(
    const float* __restrict__ x,     // [N, 3]
    const float* __restrict__ emb,   // [16, 2^19, 2]
    float* __restrict__ out,         // [N, 32]
    int n_points) {
  const int tid   = blockIdx.x * blockDim.x + threadIdx.x;
  const int level = tid & (kLevels - 1);
  const int p     = tid >> 4;  // log2(kLevels)
  if (p >= n_points) return;

  const float* xp = x + (size_t)p * 3;

  // Warm the streamed x read-ahead (one lane per point-group issues it).
  // Lowers to global_prefetch_b8 on gfx1250; speculative, no side effects.
  if (level == 0) {
    __builtin_prefetch(xp + 3 * 8192, 0, 0);
  }

  // res = 16 * 2^level, exact in f32. Reference computes pos in f32 too,
  // so px/floor/frac below are bit-identical to the reference.
  const float res = (float)(16u << level);
  const float px = xp[0] * res;
  const float py = xp[1] * res;
  const float pz = xp[2] * res;
  const float fx = floorf(px);
  const float fy = floorf(py);
  const float fz = floorf(pz);
  const float tx = px - fx;
  const float ty = py - fy;
  const float tz = pz - fz;

  // idx in [0, 524289] -> fits uint32. Only low 19 bits of idx*prime matter,
  // so 32-bit wraparound multiplies match the reference's int64 hash exactly.
  const unsigned ix = (unsigned)(int)fx;
  const unsigned iy = (unsigned)(int)fy;
  const unsigned iz = (unsigned)(int)fz;
  const unsigned a0 = ix * kP0, b0 = a0 + kP0;  // (ix+1)*P0 mod 2^32
  const unsigned a1 = iy * kP1, b1 = a1 + kP1;
  const unsigned a2 = iz * kP2, b2 = a2 + kP2;

  // Per-level table viewed as float2[2^19]; offsets are 8B-aligned.
  const float2* __restrict__ tab =
      reinterpret_cast<const float2*>(emb) + (size_t)level * kHashSize;

  // Issue all 8 gathers before consuming any -> 8 loads in flight per thread.
  float2 f[8];
#pragma unroll
  for (int c = 0; c < 8; ++c) {
    const unsigned h = ((c & 1) ? b0 : a0) ^
                       ((c & 2) ? b1 : a1) ^
                       ((c & 4) ? b2 : a2);
    f[c] = tab[h & kHashMask];
  }

  const float wx0 = 1.0f - tx, wx1 = tx;
  const float wy0 = 1.0f - ty, wy1 = ty;
  const float wz0 = 1.0f - tz, wz1 = tz;

  float s0 = 0.0f, s1 = 0.0f;
#pragma unroll
  for (int c = 0; c < 8; ++c) {
    const float w = ((c & 1) ? wx1 : wx0) *
                    ((c & 2) ? wy1 : wy0) *
                    ((c & 4) ? wz1 : wz0);
    s0 = fmaf(w, f[c].x, s0);
    s1 = fmaf(w, f[c].y, s1);
  }

  // out[p, 2*level : 2*level+2]; a wave (32 lanes) writes 256 contiguous bytes.
  float2* __restrict__ op =
      reinterpret_cast<float2*>(out + (size_t)p * (kLevels * kFeats)) + level;
  *op = make_float2(s0, s1);
}

extern "C" void kernel_launch(void* const* d_in, const int* in_sizes, int n_in,
                              void* d_out, int out_size, void* d_ws, size_t ws_size,
                              hipStream_t stream) {
  const float* x   = (const float*)d_in[0];   // [N, 3] float32
  const float* emb = (const float*)d_in[1];   // [16, 2^19, 2] float32
  float* out       = (float*)d_out;           // [N, 32] float32

  const int n_points = in_sizes[0] / 3;
  const long long total = (long long)n_points * kLevels;
  const int block = 256;
  const int grid  = (int)((total + block - 1) / block);

  hashgrid_encode_kernel<<<grid, block, 0, stream>>>(x, emb, out, n_points);
}